// RVQ_77807627534983
// MI455X (gfx1250) — compile-verified
//
#include <hip/hip_runtime.h>

typedef __attribute__((ext_vector_type(2))) float v2f;
typedef __attribute__((ext_vector_type(4))) float v4f;
typedef __attribute__((ext_vector_type(8))) float v8f;

#define NROWS (32 * 4096)   // B*T
#define MELC  128
#define DDIM  64
#define KCODE 128

#define MEL_STRIDE 132      // 128 + 4: (4*col + k) % 64 distinct across lanes
#define Z_STRIDE   68       // 64 + 4

__device__ __forceinline__ v8f wmma4(v2f a, v2f b, v8f c) {
    // D = A(16x4,f32) * B(4x16,f32) + C(16x16,f32)
    return __builtin_amdgcn_wmma_f32_16x16x4_f32(
        /*neg_a=*/false, a, /*neg_b=*/false, b,
        /*c_mod=*/(short)0, c, /*reuse_a=*/false, /*reuse_b=*/false);
}

// ---------------------------------------------------------------------------
// Precompute (tiny): M01 = cb0@cb1^T, P0 = cb0@w_out^T, P1 = cb1@w_out^T,
// c2_* = ||code||^2, plus K-pair-major packed copies of w_in/cb0/cb1 so that
// WMMA B-fragments become fully coalesced b64 loads (lanes 0-15 read 128
// contiguous bytes of one K-pair row, lanes 16-31 the next K-pair row).
// ---------------------------------------------------------------------------
__global__ void rvq_precompute(const float* __restrict__ cb0,
                               const float* __restrict__ cb1,
                               const float* __restrict__ w_out,
                               const float* __restrict__ w_in,
                               float* __restrict__ M01, float* __restrict__ P0,
                               float* __restrict__ P1, float* __restrict__ c20,
                               float* __restrict__ c21,
                               v2f* __restrict__ w_in_pk,   // [64][64]  kp-major
                               v2f* __restrict__ cb0_pk,    // [32][128] kp-major
                               v2f* __restrict__ cb1_pk) {  // [32][128] kp-major
    const int i = blockIdx.x;   // 0..127
    const int j = threadIdx.x;  // 0..127
    const float* a0 = cb0 + i * DDIM;
    const float* a1 = cb1 + i * DDIM;
    const float* b1 = cb1 + j * DDIM;
    const float* wr = w_out + j * DDIM;
    float s01 = 0.f, sp0 = 0.f, sp1 = 0.f;
#pragma unroll 8
    for (int k = 0; k < DDIM; ++k) {
        float x0 = a0[k], x1 = a1[k], wv = wr[k];
        s01 += x0 * b1[k];
        sp0 += x0 * wv;
        sp1 += x1 * wv;
    }
    M01[i * KCODE + j] = s01;
    P0[i * MELC + j] = sp0;
    P1[i * MELC + j] = sp1;
    if (i == 0) {
        const float* r = cb0 + j * DDIM;
        float s = 0.f;
        for (int k = 0; k < DDIM; ++k) s += r[k] * r[k];
        c20[j] = s;
    }
    if (i == 1) {
        const float* r = cb1 + j * DDIM;
        float s = 0.f;
        for (int k = 0; k < DDIM; ++k) s += r[k] * r[k];
        c21[j] = s;
    }
    // K-pair packing. B[k][n] = W[n][k]  ->  pk[kp][n] = {W[n][2kp], W[n][2kp+1]}
    if (i < 64 && j < 64) {
        v2f p;
        p.x = w_in[j * MELC + 2 * i];
        p.y = w_in[j * MELC + 2 * i + 1];
        w_in_pk[i * 64 + j] = p;
    }
    if (i < 32) {
        v2f p;
        p.x = cb0[j * DDIM + 2 * i];
        p.y = cb0[j * DDIM + 2 * i + 1];
        cb0_pk[i * KCODE + j] = p;
    } else if (i < 64) {
        const int kp = i - 32;
        v2f p;
        p.x = cb1[j * DDIM + 2 * kp];
        p.y = cb1[j * DDIM + 2 * kp + 1];
        cb1_pk[kp * KCODE + j] = p;
    }
}

// ---------------------------------------------------------------------------
// Main fused kernel: 8 waves/block, each wave owns 16 rows.
//   stage mel tile -> LDS (coalesced b128 in, conflict-free b64 A-frags out)
//   z  = mel @ w_in^T + b_in                  (WMMA f32 16x16x4, K=128)
//   G0 = z @ cb0^T  -> argmin(c2_0 - 2G0)             = i0
//   G1 = z @ cb1^T  -> argmin(c2_1 + 2*M01[i0] - 2G1) = i1
//   out = P0[i0] + P1[i1] + b_out             (gather, no GEMM)
// ---------------------------------------------------------------------------
__global__ __launch_bounds__(256) void rvq_main(
    const float* __restrict__ mel, const float* __restrict__ b_in,
    const v2f* __restrict__ w_in_pk, const v2f* __restrict__ cb0_pk,
    const v2f* __restrict__ cb1_pk, const float* __restrict__ b_out,
    const float* __restrict__ M01, const float* __restrict__ P0,
    const float* __restrict__ P1, const float* __restrict__ c2_0,
    const float* __restrict__ c2_1, float* __restrict__ out) {
    __shared__ float lds_mel[8][16 * MEL_STRIDE];  // 66 KB
    __shared__ float lds_z[8][16 * Z_STRIDE];      // 34 KB
    __shared__ int lds_i0[8][16];
    __shared__ int lds_i1[8][16];

    const int w = threadIdx.x >> 5;     // wave in block
    const int lane = threadIdx.x & 31;
    const int half = lane >> 4;         // selects K pair within WMMA frag
    const int col = lane & 15;          // M (for A) / N (for B,C,D)
    const int r0 = blockIdx.x * 128 + w * 16;

    // ---- stage this wave's 16x128 mel tile (contiguous 8KB) into LDS ----
    {
        const float* src = mel + (size_t)r0 * MELC;
#pragma unroll
        for (int i = 0; i < 16; ++i) {
            v4f vv = *(const v4f*)(src + i * MELC + lane * 4);
            *(v4f*)(&lds_mel[w][i * MEL_STRIDE + lane * 4]) = vv;
        }
    }

    // ---------------- GEMM1: z = mel(16x128) @ w_in^T + b_in ----------------
    v8f zacc[4];
#pragma unroll
    for (int nt = 0; nt < 4; ++nt) zacc[nt] = {};

    const float* melA = &lds_mel[w][col * MEL_STRIDE + 2 * half];
#pragma unroll 4
    for (int k0 = 0; k0 < MELC; k0 += 4) {
        v2f a = *(const v2f*)(melA + k0);
        const v2f* brow = w_in_pk + ((k0 >> 1) + half) * 64 + col;
#pragma unroll
        for (int nt = 0; nt < 4; ++nt) {
            v2f b = brow[nt * 16];
            zacc[nt] = wmma4(a, b, zacc[nt]);
        }
    }
    // bias + stage z into LDS (row-major, padded stride)
#pragma unroll
    for (int nt = 0; nt < 4; ++nt) {
        float bv = b_in[nt * 16 + col];
#pragma unroll
        for (int v = 0; v < 8; ++v)
            lds_z[w][(v + 8 * half) * Z_STRIDE + nt * 16 + col] = zacc[nt][v] + bv;
    }
    __syncthreads();

    const float* zrow = &lds_z[w][col * Z_STRIDE + 2 * half];

    // ---------------- codebook 0: G0 = z @ cb0^T, argmin ----------------
    {
        v8f g[8];
#pragma unroll
        for (int t = 0; t < 8; ++t) g[t] = {};
#pragma unroll 2
        for (int k0 = 0; k0 < DDIM; k0 += 4) {
            v2f a = *(const v2f*)(zrow + k0);
            const v2f* brow = cb0_pk + ((k0 >> 1) + half) * KCODE + col;
#pragma unroll
            for (int t = 0; t < 8; ++t) {
                v2f b = brow[t * 16];
                g[t] = wmma4(a, b, g[t]);
            }
        }
        float minv[8];
        int mini[8];
#pragma unroll
        for (int v = 0; v < 8; ++v) { minv[v] = 3.4e38f; mini[v] = 0; }
#pragma unroll
        for (int t = 0; t < 8; ++t) {
            const int c = t * 16 + col;
            const float cc = c2_0[c];
#pragma unroll
            for (int v = 0; v < 8; ++v) {
                float s = cc - 2.0f * g[t][v];
                if (s < minv[v]) { minv[v] = s; mini[v] = c; }
            }
        }
#pragma unroll
        for (int m = 8; m >= 1; m >>= 1) {
#pragma unroll
            for (int v = 0; v < 8; ++v) {
                float ov = __shfl_xor(minv[v], m, 32);
                int oi = __shfl_xor(mini[v], m, 32);
                if (ov < minv[v] || (ov == minv[v] && oi < mini[v])) {
                    minv[v] = ov;
                    mini[v] = oi;
                }
            }
        }
        if (col == 0) {
#pragma unroll
            for (int v = 0; v < 8; ++v) lds_i0[w][v + 8 * half] = mini[v];
        }
    }
    __syncthreads();

    // ---------------- codebook 1: G1 = z @ cb1^T, corrected argmin ----------
    {
        int i0v[8];
#pragma unroll
        for (int v = 0; v < 8; ++v) i0v[v] = lds_i0[w][v + 8 * half];

        v8f g[8];
#pragma unroll
        for (int t = 0; t < 8; ++t) g[t] = {};
#pragma unroll 2
        for (int k0 = 0; k0 < DDIM; k0 += 4) {
            v2f a = *(const v2f*)(zrow + k0);
            const v2f* brow = cb1_pk + ((k0 >> 1) + half) * KCODE + col;
#pragma unroll
            for (int t = 0; t < 8; ++t) {
                v2f b = brow[t * 16];
                g[t] = wmma4(a, b, g[t]);
            }
        }
        float minv[8];
        int mini[8];
#pragma unroll
        for (int v = 0; v < 8; ++v) { minv[v] = 3.4e38f; mini[v] = 0; }
#pragma unroll
        for (int t = 0; t < 8; ++t) {
            const int c = t * 16 + col;
            const float cc = c2_1[c];
#pragma unroll
            for (int v = 0; v < 8; ++v) {
                // ||r - c||^2 argmin with r = z - cb0[i0]:
                // c2_1[c] - 2*(z.c - cb0[i0].c) = c2_1[c] + 2*M01[i0][c] - 2*G1
                float s = cc + 2.0f * M01[i0v[v] * KCODE + c] - 2.0f * g[t][v];
                if (s < minv[v]) { minv[v] = s; mini[v] = c; }
            }
        }
#pragma unroll
        for (int m = 8; m >= 1; m >>= 1) {
#pragma unroll
            for (int v = 0; v < 8; ++v) {
                float ov = __shfl_xor(minv[v], m, 32);
                int oi = __shfl_xor(mini[v], m, 32);
                if (ov < minv[v] || (ov == minv[v] && oi < mini[v])) {
                    minv[v] = ov;
                    mini[v] = oi;
                }
            }
        }
        if (col == 0) {
#pragma unroll
            for (int v = 0; v < 8; ++v) lds_i1[w][v + 8 * half] = mini[v];
        }
    }
    __syncthreads();

    // ---------------- decode: out = P0[i0] + P1[i1] + b_out ----------------
    const v4f bo = *(const v4f*)(b_out + lane * 4);
#pragma unroll 4
    for (int m = 0; m < 16; ++m) {
        const int i0m = lds_i0[w][m];
        const int i1m = lds_i1[w][m];
        v4f p0 = *(const v4f*)(P0 + i0m * MELC + lane * 4);
        v4f p1 = *(const v4f*)(P1 + i1m * MELC + lane * 4);
        *(v4f*)(out + (size_t)(r0 + m) * MELC + lane * 4) = p0 + p1 + bo;
    }
}

extern "C" void kernel_launch(void* const* d_in, const int* in_sizes, int n_in,
                              void* d_out, int out_size, void* d_ws, size_t ws_size,
                              hipStream_t stream) {
    (void)in_sizes; (void)n_in; (void)out_size; (void)ws_size;
    const float* mel   = (const float*)d_in[0];
    const float* w_in  = (const float*)d_in[1];
    const float* b_in  = (const float*)d_in[2];
    const float* cb0   = (const float*)d_in[3];
    const float* cb1   = (const float*)d_in[4];
    const float* w_out = (const float*)d_in[5];
    const float* b_out = (const float*)d_in[6];
    float* out = (float*)d_out;

    float* ws  = (float*)d_ws;
    float* M01 = ws;                 // 16384 floats
    float* P0  = ws + 16384;         // 16384
    float* P1  = ws + 32768;         // 16384
    float* c20 = ws + 49152;         // 128
    float* c21 = ws + 49280;         // 128
    v2f* w_in_pk = (v2f*)(ws + 49408);           // 4096 v2f = 8192 floats
    v2f* cb0_pk  = (v2f*)(ws + 49408 + 8192);    // 4096 v2f
    v2f* cb1_pk  = (v2f*)(ws + 49408 + 16384);   // 4096 v2f

    rvq_precompute<<<KCODE, KCODE, 0, stream>>>(cb0, cb1, w_out, w_in, M01, P0,
                                                P1, c20, c21, w_in_pk, cb0_pk,
                                                cb1_pk);
    rvq_main<<<NROWS / 128, 256, 0, stream>>>(mel, b_in, w_in_pk, cb0_pk,
                                              cb1_pk, b_out, M01, P0, P1, c20,
                                              c21, out);
}